// Topt_CBAM_22866405883932
// MI455X (gfx1250) — compile-verified
//
#include <hip/hip_runtime.h>
#include <math.h>

#define B_DIM 32
#define C_DIM 512
#define R_DIM 32
#define H_DIM 56
#define W_DIM 56
#define HW    3136
#define K_TOP 314          // round(3136 * 0.1)
#define NT    256
#define VPT   13           // ceil(3136/256)
#define BN_EPS 1e-5f

typedef float v2f __attribute__((ext_vector_type(2)));
typedef float v8f __attribute__((ext_vector_type(8)));

// ---------------- wave32-correct block reductions (8 waves / 256 thr) -------
__device__ __forceinline__ int blockReduceSumI(int v, int* red, int tid) {
  #pragma unroll
  for (int off = 16; off; off >>= 1) v += __shfl_xor(v, off, 32);
  if ((tid & 31) == 0) red[tid >> 5] = v;
  __syncthreads();
  int t = 0;
  #pragma unroll
  for (int i = 0; i < 8; ++i) t += red[i];
  __syncthreads();
  return t;
}

__device__ __forceinline__ float blockReduceSumF(float v, float* red, int tid) {
  #pragma unroll
  for (int off = 16; off; off >>= 1) v += __shfl_xor(v, off, 32);
  if ((tid & 31) == 0) red[tid >> 5] = v;
  __syncthreads();
  float t = 0.f;
  #pragma unroll
  for (int i = 0; i < 8; ++i) t += red[i];
  __syncthreads();
  return t;
}

__device__ __forceinline__ float blockReduceMaxF(float v, float* red, int tid) {
  #pragma unroll
  for (int off = 16; off; off >>= 1) v = fmaxf(v, __shfl_xor(v, off, 32));
  if ((tid & 31) == 0) red[tid >> 5] = v;
  __syncthreads();
  float t = red[0];
  #pragma unroll
  for (int i = 1; i < 8; ++i) t = fmaxf(t, red[i]);
  __syncthreads();
  return t;
}

// ---------------- Kernel 1: exact top-314 mean + max per (b,c) row ----------
// MSB radix-select on order-preserving uint key; exact tie handling matches
// lax.top_k:  sum_top = sum(values > thr) + want * thr.
__global__ void topt_pool_kernel(const float* __restrict__ x,
                                 float* __restrict__ pool_avg,
                                 float* __restrict__ pool_max) {
  __shared__ int   redI[8];
  __shared__ float redF[8];
  const int tid = threadIdx.x;
  const int row = blockIdx.x;                 // b*C + c
  const float* xr = x + (size_t)row * HW;

  float    v[VPT];
  unsigned key[VPT];
  float lmax = -3.402823466e+38f;
  #pragma unroll
  for (int i = 0; i < VPT; ++i) {
    int idx = tid + i * NT;
    if (idx < HW) {
      float f = xr[idx];
      v[i] = f;
      unsigned u = __float_as_uint(f);
      key[i] = (u & 0x80000000u) ? ~u : (u | 0x80000000u);  // monotone key, key==0 impossible for finite x
      lmax = fmaxf(lmax, f);
    } else { v[i] = 0.f; key[i] = 0u; }
  }
  float gmax = blockReduceMaxF(lmax, redF, tid);

  unsigned want = K_TOP;     // values still needed among current candidates
  unsigned prefix = 0u;      // high bits [31:bit+1] of the 314th-largest key
  for (int bit = 31; bit >= 0; --bit) {
    int cnt = 0;
    #pragma unroll
    for (int i = 0; i < VPT; ++i) {
      unsigned kk = key[i];
      bool cand = (bit == 31) ? (kk != 0u) : ((kk >> (bit + 1)) == prefix);
      cnt += (cand && ((kk >> bit) & 1u)) ? 1 : 0;
    }
    int total = blockReduceSumI(cnt, redI, tid);
    if ((unsigned)total >= want) prefix = (prefix << 1) | 1u;
    else { want -= (unsigned)total; prefix <<= 1; }
  }

  float sgt = 0.f;                            // sum of strictly-greater values
  #pragma unroll
  for (int i = 0; i < VPT; ++i) if (key[i] > prefix) sgt += v[i];
  float tot = blockReduceSumF(sgt, redF, tid);

  if (tid == 0) {
    unsigned u = (prefix & 0x80000000u) ? (prefix & 0x7FFFFFFFu) : ~prefix;
    float thr = __uint_as_float(u);
    float s = tot + (float)want * thr;
    pool_avg[row] = s * (1.0f / (float)K_TOP);
    pool_max[row] = gmax;                     // max of top-k == overall max
  }
}

// ---------------- Kernel 2: MLP gate via V_WMMA_F32_16X16X4_F32 -------------
// Phase 1: h1[64x32] = relu([avg;max][64x512] @ w1^T + b1)   (A 16x4, B 4x16)
// Phase 2: att[32x512] = (h1a+h1m) @ w2^T + 2*b2 ; sig_att = sigmoid(att)
__global__ void mlp_wmma_kernel(const float* __restrict__ pool_avg,
                                const float* __restrict__ pool_max,
                                const float* __restrict__ w1,   // [R=32, C=512]
                                const float* __restrict__ b1,   // [32]
                                const float* __restrict__ w2,   // [C=512, R=32]
                                const float* __restrict__ b2,   // [512]
                                float* __restrict__ sig_att) {  // [32, 512]
  __shared__ float h1s[64 * 32];
  const int tid  = threadIdx.x;
  const int wid  = tid >> 5;      // 8 waves
  const int lane = tid & 31;
  const int lrow = lane & 15;     // M/N within tile
  const int lg   = lane >> 4;     // K-pair group (ISA 32-bit A/B layout)

  // --- phase 1: one 16x16 tile per wave; mt 0..3 over [avg;max], nt 0..1 ---
  {
    const int mt = wid >> 1;
    const int nt = wid & 1;
    const float* pool = (mt < 2) ? pool_avg : pool_max;
    const int mrow = ((mt & 1) * 16) + lrow;      // row within [0,32)
    const int nrow = nt * 16 + lrow;              // w1 row (R index)
    v8f acc = {};
    for (int k0 = 0; k0 < C_DIM; k0 += 4) {
      const int ka = k0 + lg * 2;
      v2f a, bf;
      a.x  = pool[mrow * C_DIM + ka];
      a.y  = pool[mrow * C_DIM + ka + 1];
      bf.x = w1[nrow * C_DIM + ka];
      bf.y = w1[nrow * C_DIM + ka + 1];
      acc = __builtin_amdgcn_wmma_f32_16x16x4_f32(
          false, a, false, bf, (short)0, acc, false, false);
    }
    const float bias = b1[nt * 16 + lrow];
    #pragma unroll
    for (int vv = 0; vv < 8; ++vv) {
      const int m_global = mt * 16 + vv + 8 * lg;     // D layout: M = v + 8*(lane/16)
      const float hv = acc[vv] + bias;
      h1s[m_global * 32 + (nt * 16 + lrow)] = hv > 0.f ? hv : 0.f;
    }
  }
  __syncthreads();

  // --- phase 2: 64 tiles (mt2: 2 batch tiles, nt2: 32 channel tiles) ---
  for (int t = wid; t < 64; t += 8) {
    const int mt2 = t & 1;
    const int nt2 = t >> 1;
    const int m2 = mt2 * 16 + lrow;                   // batch row
    const int n2 = nt2 * 16 + lrow;                   // output channel (w2 row)
    v8f acc = {};
    for (int k0 = 0; k0 < 32; k0 += 4) {
      const int ka = k0 + lg * 2;
      v2f a, bf;
      a.x  = h1s[m2 * 32 + ka]     + h1s[(m2 + 32) * 32 + ka];
      a.y  = h1s[m2 * 32 + ka + 1] + h1s[(m2 + 32) * 32 + ka + 1];
      bf.x = w2[n2 * 32 + ka];
      bf.y = w2[n2 * 32 + ka + 1];
      acc = __builtin_amdgcn_wmma_f32_16x16x4_f32(
          false, a, false, bf, (short)0, acc, false, false);
    }
    const float bias2 = 2.0f * b2[nt2 * 16 + lrow];
    #pragma unroll
    for (int vv = 0; vv < 8; ++vv) {
      const int bb = mt2 * 16 + vv + 8 * lg;
      const float att = acc[vv] + bias2;
      sig_att[bb * C_DIM + (nt2 * 16 + lrow)] = 1.0f / (1.0f + expf(-att));
    }
  }
}

// ---------------- Kernel 3: channel max / mean of x * sig_att ---------------
__global__ void channel_reduce_kernel(const float* __restrict__ x,
                                      const float* __restrict__ sig_att,
                                      float* __restrict__ cp_max,
                                      float* __restrict__ cp_mean) {
  __shared__ float satt[C_DIM];
  const int tid = threadIdx.x;
  const int b = blockIdx.y;
  for (int c = tid; c < C_DIM; c += NT) satt[c] = sig_att[b * C_DIM + c];
  __syncthreads();
  const int pix = blockIdx.x * NT + tid;
  if (pix >= HW) return;
  const float* xb = x + (size_t)b * C_DIM * HW + pix;
  float mx = -3.402823466e+38f, sm = 0.f;
  for (int c = 0; c < C_DIM; ++c) {
    if ((c & 7) == 0 && c + 8 < C_DIM)
      __builtin_prefetch(xb + (size_t)(c + 8) * HW, 0, 1);   // global_prefetch_b8
    const float f = xb[(size_t)c * HW] * satt[c];
    mx = fmaxf(mx, f);
    sm += f;
  }
  cp_max[b * HW + pix]  = mx;
  cp_mean[b * HW + pix] = sm * (1.0f / (float)C_DIM);
}

// ---------------- Kernel 4: 7x7 conv + BN + sigmoid -------------------------
__global__ void spatial_gate_kernel(const float* __restrict__ cp_max,
                                    const float* __restrict__ cp_mean,
                                    const float* __restrict__ conv_w, // [1,2,7,7]
                                    const float* __restrict__ bn_gamma,
                                    const float* __restrict__ bn_beta,
                                    const float* __restrict__ bn_mean,
                                    const float* __restrict__ bn_var,
                                    float* __restrict__ sgate) {
  __shared__ float wmx[49], wme[49];
  const int tid = threadIdx.x;
  if (tid < 49) { wmx[tid] = conv_w[tid]; wme[tid] = conv_w[49 + tid]; }
  __syncthreads();
  const int pix = blockIdx.x * NT + tid;
  if (pix >= HW) return;
  const int b = blockIdx.y;
  const int hy = pix / W_DIM, hx = pix % W_DIM;
  const float* pm = cp_max + b * HW;
  const float* pe = cp_mean + b * HW;
  float acc = 0.f;
  #pragma unroll
  for (int ky = 0; ky < 7; ++ky) {
    const int iy = hy + ky - 3;
    if (iy < 0 || iy >= H_DIM) continue;
    #pragma unroll
    for (int kx = 0; kx < 7; ++kx) {
      const int ix = hx + kx - 3;
      if (ix < 0 || ix >= W_DIM) continue;
      const int ip = iy * W_DIM + ix;
      acc += pm[ip] * wmx[ky * 7 + kx] + pe[ip] * wme[ky * 7 + kx];
    }
  }
  const float bn = (acc - bn_mean[0]) * rsqrtf(bn_var[0] + BN_EPS) * bn_gamma[0]
                 + bn_beta[0];
  sgate[b * HW + pix] = 1.0f / (1.0f + expf(-bn));
}

// ---------------- Kernel 5: out = x * sig_att * sgate (float4 streams) ------
__global__ void final_apply_kernel(const float* __restrict__ x,
                                   const float* __restrict__ sig_att,
                                   const float* __restrict__ sgate,
                                   float* __restrict__ out) {
  const size_t i4   = (size_t)blockIdx.x * NT + threadIdx.x;
  const size_t base = i4 * 4;                       // HW%4==0 -> uniform (b,c)
  const int b  = (int)(base / ((size_t)C_DIM * HW));
  const size_t rem = base % ((size_t)C_DIM * HW);
  const int c  = (int)(rem / HW);
  const int hw = (int)(rem % HW);
  const float s = sig_att[b * C_DIM + c];
  const float4 xv = *(const float4*)(x + base);
  const float4 gv = *(const float4*)(sgate + (size_t)b * HW + hw);
  float4 o;
  o.x = xv.x * s * gv.x;
  o.y = xv.y * s * gv.y;
  o.z = xv.z * s * gv.z;
  o.w = xv.w * s * gv.w;
  *(float4*)(out + base) = o;
}

// ---------------- host launcher --------------------------------------------
extern "C" void kernel_launch(void* const* d_in, const int* in_sizes, int n_in,
                              void* d_out, int out_size, void* d_ws, size_t ws_size,
                              hipStream_t stream) {
  const float* x        = (const float*)d_in[0];
  const float* w1       = (const float*)d_in[1];
  const float* b1       = (const float*)d_in[2];
  const float* w2       = (const float*)d_in[3];
  const float* b2       = (const float*)d_in[4];
  const float* conv_w   = (const float*)d_in[5];
  const float* bn_gamma = (const float*)d_in[6];
  const float* bn_beta  = (const float*)d_in[7];
  const float* bn_mean  = (const float*)d_in[8];
  const float* bn_var   = (const float*)d_in[9];

  float* ws       = (float*)d_ws;
  float* pool_avg = ws;                 // 16384
  float* pool_max = ws + 16384;         // 16384
  float* sig_att  = ws + 32768;         // 16384
  float* cp_max   = ws + 49152;         // 100352
  float* cp_mean  = ws + 149504;        // 100352
  float* sgate    = ws + 249856;        // 100352  (total ~1.37 MB)

  topt_pool_kernel<<<B_DIM * C_DIM, NT, 0, stream>>>(x, pool_avg, pool_max);
  mlp_wmma_kernel<<<1, NT, 0, stream>>>(pool_avg, pool_max, w1, b1, w2, b2, sig_att);
  channel_reduce_kernel<<<dim3((HW + NT - 1) / NT, B_DIM), NT, 0, stream>>>(
      x, sig_att, cp_max, cp_mean);
  spatial_gate_kernel<<<dim3((HW + NT - 1) / NT, B_DIM), NT, 0, stream>>>(
      cp_max, cp_mean, conv_w, bn_gamma, bn_beta, bn_mean, bn_var, sgate);
  final_apply_kernel<<<(B_DIM * C_DIM * HW / 4) / NT, NT, 0, stream>>>(
      x, sig_att, sgate, (float*)d_out);
}